// Attention_14602888806957
// MI455X (gfx1250) — compile-verified
//
#include <hip/hip_runtime.h>
#include <math.h>

// ---------------- problem constants ----------------
constexpr int B = 32;
constexpr int L = 2048;
constexpr int H = 1024;
constexpr int S = 8;            // L-splits per batch (flash phase)
constexpr int ROWS = L / S;     // 256 rows per block
constexpr int CH = 32;          // chunk rows for online softmax
constexpr int NCH = ROWS / CH;  // 8 chunks
constexpr int CHUNK_FLOATS = CH * H;   // 32768 floats = 128 KB

// d_out layout (floats): out[B*H] | attn[B*L] | coverage_new[B*L]
constexpr int ATTN_OFF = B * H;            // 32768
constexpr int COV_OFF  = ATTN_OFF + B * L; // 98304

// workspace layout (floats)
constexpr int CW_OFF   = 0;                       // cov_w . W_w
constexpr int CB_OFF   = 1;                       // cov_b . W_w + W_b
constexpr int SOUT_OFF = 2;                       // s_out[b] = output[b] . W_w  (32)
constexpr int PM_OFF   = 64;                      // per-split running max   [B*S]
constexpr int PSUM_OFF = PM_OFF + B * S;          // per-split running sum   [B*S]
constexpr int PMIX_OFF = PSUM_OFF + B * S;        // per-split mix           [B*S*H]
constexpr int COMB_OFF = PMIX_OFF + B * S * H;    // combined [B, 2H]
// total = COMB_OFF + B*2*H floats  (~1.31 MB)

typedef float v2f_t __attribute__((ext_vector_type(2)));
typedef float v8f_t __attribute__((ext_vector_type(8)));

// ---------------------------------------------------------------------------
// CDNA5 async global->LDS helpers (ASYNCcnt path, CDNA5 ISA)
// ---------------------------------------------------------------------------
// Issue one wave-wide 128-bit async copy: each lane copies 16 bytes from the global address in ga to the LDS address in la.
__device__ __forceinline__ void async_b128(unsigned la, unsigned long long ga) {
    asm volatile("global_load_async_to_lds_b128 %0, %1, off"
                 :: "v"(la), "v"(ga) : "memory");
}
__device__ __forceinline__ void wait_async_le(int) = delete;
__device__ __forceinline__ void wait_async_32() {
    asm volatile("s_wait_asynccnt 0x20" ::: "memory");
}
__device__ __forceinline__ void wait_async_0() {
    asm volatile("s_wait_asynccnt 0x0" ::: "memory");
}

// Stage one 32x1024 f32 chunk (128 KB) into LDS: 32 b128 issues per thread.
__device__ __forceinline__ void async_load_chunk(const float* __restrict__ gsrc,
                                                 float* __restrict__ ldst, int tid) {
    // generic shared pointer -> LDS byte address = low 32 bits (ISA 10.2)
    const unsigned lbase = (unsigned)(size_t)ldst;
    const unsigned long long gbase = (unsigned long long)(size_t)gsrc;
#pragma unroll
    for (int i = 0; i < 32; ++i) {
        const unsigned byte = (unsigned)(i * 256 + tid) * 16u;   // float4 granularity
        async_b128(lbase + byte, gbase + byte);
    }
}

// ---------------------------------------------------------------------------
// Kernel 1: scalar precomputation.
// blocks 0..31 : s_out[b] = dot(output[b], W_w)
// block  32    : cw = dot(cov_w, W_w); cb = dot(cov_b, W_w) + W_b
// ---------------------------------------------------------------------------
__global__ __launch_bounds__(256)
void k_scalars(const float* __restrict__ output, const float* __restrict__ Ww,
               const float* __restrict__ Wb, const float* __restrict__ covw,
               const float* __restrict__ covb, float* __restrict__ ws) {
    __shared__ float r1[256];
    __shared__ float r2[256];
    const int b = blockIdx.x;
    const int t = threadIdx.x;
    float a1 = 0.f, a2 = 0.f;
    if (b < B) {
        for (int k = t; k < H; k += 256) a1 += output[b * H + k] * Ww[k];
    } else {
        for (int k = t; k < H; k += 256) {
            const float w = Ww[k];
            a1 += covw[k] * w;
            a2 += covb[k] * w;
        }
    }
    r1[t] = a1; r2[t] = a2;
    __syncthreads();
    for (int s2 = 128; s2 > 0; s2 >>= 1) {
        if (t < s2) { r1[t] += r1[t + s2]; r2[t] += r2[t + s2]; }
        __syncthreads();
    }
    if (t == 0) {
        if (b < B) ws[SOUT_OFF + b] = r1[0];
        else { ws[CW_OFF] = r1[0]; ws[CB_OFF] = r2[0] + Wb[0]; }
    }
}

// ---------------------------------------------------------------------------
// Kernel 2: fused score + online softmax + mix accumulation, single HBM pass
// over context. One block per (batch, L-split). Each 128 KB chunk is streamed
// into a double-buffered LDS tile with GLOBAL_LOAD_ASYNC_TO_LDS_B128 while the
// previous chunk is scored/mixed out of LDS. Raw scores are parked in the attn
// region of d_out for kernel 3 to normalize.
// Dynamic LDS: 2*128KB chunk buffers + Ww + softmax state (~261.5 KB < 320 KB).
// ---------------------------------------------------------------------------
constexpr int FLASH_SMEM_FLOATS = 2 * CHUNK_FLOATS + H + ROWS + CH + 4;
constexpr size_t FLASH_SMEM_BYTES = (size_t)FLASH_SMEM_FLOATS * sizeof(float);

__global__ __launch_bounds__(256)
void k_flash(const float* __restrict__ ctx, const float* __restrict__ coverage,
             const float* __restrict__ Ww, float* __restrict__ ws,
             float* __restrict__ out) {
    extern __shared__ float smem[];
    float* bufs = smem;                        // [2][CH*H]
    float* sWw  = smem + 2 * CHUNK_FLOATS;     // [H]
    float* cs   = sWw + H;                     // [ROWS]
    float* pbuf = cs + ROWS;                   // [CH]
    float* shs  = pbuf + CH;                   // [0]=m  [1]=l  [2]=scale

    const int b   = blockIdx.x / S;
    const int s   = blockIdx.x % S;
    const int tid = threadIdx.x;
    const int lane = tid & 31;
    const int w    = tid >> 5;

    const float* ctxb = ctx + (size_t)b * L * H;
    const int l0 = s * ROWS;
    const int h0 = tid * 4;

    // kick off chunk 0 before anything else, then stage Ww
    async_load_chunk(ctxb + (size_t)l0 * H, bufs, tid);
    for (int k = tid; k < H; k += 256) sWw[k] = Ww[k];
    if (tid == 0) { shs[0] = -INFINITY; shs[1] = 0.f; }

    const float cw    = ws[CW_OFF];
    const float cb    = ws[CB_OFF];
    const float s_out = ws[SOUT_OFF + b];

    float4 mix = make_float4(0.f, 0.f, 0.f, 0.f);

    for (int c = 0; c < NCH; ++c) {
        float* buf = bufs + (c & 1) * CHUNK_FLOATS;
        // stream next chunk into the other buffer; wait for this chunk only
        // (async loads complete in order -> waiting ASYNCcnt<=32 retires the
        //  32 ops of chunk c while chunk c+1 stays in flight)
        if (c + 1 < NCH) {
            async_load_chunk(ctxb + (size_t)(l0 + (c + 1) * CH) * H,
                             bufs + ((c + 1) & 1) * CHUNK_FLOATS, tid);
            wait_async_32();
        } else {
            wait_async_0();
        }
        __syncthreads();   // every wave's slice of chunk c is in LDS

        // ---- scores for rows [c*32, c*32+32) out of LDS: wave w owns 4 rows
        for (int j = 0; j < 4; ++j) {
            const int rl = c * CH + w * 4 + j;
            const int l  = l0 + rl;
            const float4* row4 = (const float4*)(buf + (size_t)(w * 4 + j) * H);
            float acc = 0.f;
#pragma unroll
            for (int i = 0; i < 8; ++i) {
                const float4 v  = row4[lane + 32 * i];
                const float4 wv = ((const float4*)sWw)[lane + 32 * i];
                acc += v.x * wv.x + v.y * wv.y + v.z * wv.z + v.w * wv.w;
            }
            for (int off = 16; off > 0; off >>= 1)
                acc += __shfl_xor(acc, off, 32);
            if (lane == 0) {
                const float sc = acc + s_out + coverage[b * L + l] * cw + cb;
                cs[rl] = sc;
                out[ATTN_OFF + b * L + l] = sc;   // raw score, normalized later
            }
        }
        __syncthreads();

        // ---- online softmax update (wave 0) ----
        if (w == 0) {
            const float v = cs[c * CH + lane];
            float cm = v;
            for (int off = 16; off > 0; off >>= 1)
                cm = fmaxf(cm, __shfl_xor(cm, off, 32));
            const float new_m = fmaxf(shs[0], cm);
            const float p = expf(v - new_m);
            float csum = p;
            for (int off = 16; off > 0; off >>= 1)
                csum += __shfl_xor(csum, off, 32);
            pbuf[lane] = p;
            if (lane == 0) {
                const float scale = expf(shs[0] - new_m);
                shs[1] = shs[1] * scale + csum;
                shs[0] = new_m;
                shs[2] = scale;
            }
        }
        __syncthreads();

        // ---- rescale + accumulate mix over this chunk (pure LDS reads) ----
        const float scale = shs[2];
        mix.x *= scale; mix.y *= scale; mix.z *= scale; mix.w *= scale;
        const float* base = buf + h0;
#pragma unroll 8
        for (int j = 0; j < CH; ++j) {
            const float p = pbuf[j];
            const float4 v = *(const float4*)(base + (size_t)j * H);
            mix.x += p * v.x; mix.y += p * v.y;
            mix.z += p * v.z; mix.w += p * v.w;
        }
        __syncthreads();   // all reads done before anyone refills this buffer
    }

    const int bs = b * S + s;
    if (tid == 0) { ws[PM_OFF + bs] = shs[0]; ws[PSUM_OFF + bs] = shs[1]; }
    *(float4*)(ws + PMIX_OFF + (size_t)bs * H + h0) = mix;
}

// ---------------------------------------------------------------------------
// Kernel 3: merge the S splits per batch; emit attn, coverage_new and the
// combined = [mix | output] matrix for the final GEMM.
// ---------------------------------------------------------------------------
__global__ __launch_bounds__(256)
void k_combine(const float* __restrict__ output, const float* __restrict__ coverage,
               float* __restrict__ ws, float* __restrict__ out) {
    const int b = blockIdx.x;
    const int tid = threadIdx.x;
    __shared__ float sh_coef[S];
    __shared__ float sh_M, sh_inv;

    if (tid == 0) {
        float M = -INFINITY;
        for (int i = 0; i < S; ++i) M = fmaxf(M, ws[PM_OFF + b * S + i]);
        float Sum = 0.f;
        for (int i = 0; i < S; ++i)
            Sum += ws[PSUM_OFF + b * S + i] * expf(ws[PM_OFF + b * S + i] - M);
        const float inv = 1.f / Sum;
        for (int i = 0; i < S; ++i)
            sh_coef[i] = expf(ws[PM_OFF + b * S + i] - M) * inv;
        sh_M = M; sh_inv = inv;
    }
    __syncthreads();

    // mix -> combined[:, 0:H], output -> combined[:, H:2H]
    const int h0 = tid * 4;
    float4 acc = make_float4(0.f, 0.f, 0.f, 0.f);
    for (int i = 0; i < S; ++i) {
        const float cf = sh_coef[i];
        const float4 v = *(const float4*)(ws + PMIX_OFF + (size_t)(b * S + i) * H + h0);
        acc.x += cf * v.x; acc.y += cf * v.y; acc.z += cf * v.z; acc.w += cf * v.w;
    }
    *(float4*)(ws + COMB_OFF + (size_t)b * (2 * H) + h0) = acc;
    *(float4*)(ws + COMB_OFF + (size_t)b * (2 * H) + H + h0) =
        *(const float4*)(output + (size_t)b * H + h0);

    // normalize attn (raw scores were parked in d_out) + coverage_new
    const float M = sh_M, inv = sh_inv;
    for (int l = tid; l < L; l += 256) {
        const float a = expf(out[ATTN_OFF + b * L + l] - M) * inv;
        out[ATTN_OFF + b * L + l] = a;
        out[COV_OFF + b * L + l]  = coverage[b * L + l] + a;
    }
}

// ---------------------------------------------------------------------------
// Kernel 4: out = tanh(combined @ out_w + out_b) via V_WMMA_F32_16X16X4_F32.
// M=32 (2 tiles), N=1024 (64 tiles), K=2048. Grid (8,2) x 256 threads:
// block stages combined[m0:m0+16, kc:kc+512] in padded LDS; each wave owns one
// 16-wide N tile. A/B layouts per ISA: K0,K1 in lanes 0-15; K2,K3 in 16-31.
// ---------------------------------------------------------------------------
constexpr int KC  = 512;
constexpr int LDA = KC + 4;   // padded row stride: 4-bank rotation, 16B aligned

__global__ __launch_bounds__(256)
void k_gemm(const float* __restrict__ out_w, const float* __restrict__ out_b,
            const float* __restrict__ ws, float* __restrict__ out) {
    __shared__ float ldsA[16 * LDA];
    const int m0   = blockIdx.y * 16;
    const int tid  = threadIdx.x;
    const int lane = tid & 31;
    const int w    = tid >> 5;
    const int lr   = lane & 15;      // row within tile / column within tile
    const int half = lane >> 4;      // 0: K=0,1  1: K=2,3
    const int koff = half * 2;
    const int n0   = blockIdx.x * 128 + w * 16;
    const float* comb = ws + COMB_OFF;

    v8f_t c = {0.f, 0.f, 0.f, 0.f, 0.f, 0.f, 0.f, 0.f};

    for (int kc = 0; kc < 2 * H; kc += KC) {
        // stage A chunk (16 x 512) into LDS, float4, fully coalesced
        for (int idx = tid * 4; idx < 16 * KC; idx += 256 * 4) {
            const int r  = idx >> 9;       // /512
            const int cp = idx & (KC - 1);
            *(float4*)(ldsA + r * LDA + cp) =
                *(const float4*)(comb + (size_t)(m0 + r) * (2 * H) + kc + cp);
        }
        __syncthreads();

        const float* Arow = ldsA + lr * LDA + koff;
        const float* Bb   = out_w + (size_t)(kc + koff) * H + n0 + lr;
#pragma unroll 8
        for (int k0 = 0; k0 < KC; k0 += 4) {
            const v2f_t a = *(const v2f_t*)(Arow + k0);      // A[lr][k+koff], [k+koff+1]
            v2f_t bm;
            bm.x = Bb[(size_t)k0 * H];                       // B[k+koff]  [n0+lr]
            bm.y = Bb[(size_t)k0 * H + H];                   // B[k+koff+1][n0+lr]
            c = __builtin_amdgcn_wmma_f32_16x16x4_f32(
                    false, a, false, bm, (short)0, c, false, false);
        }
        __syncthreads();
    }

    // C layout: VGPR i -> row m0 + i + 8*half, col n0 + lr
#pragma unroll
    for (int i = 0; i < 8; ++i) {
        const int row = m0 + i + 8 * half;
        const int col = n0 + lr;
        out[(size_t)row * H + col] = tanhf(c[i] + out_b[col]);
    }
}

// ---------------------------------------------------------------------------
extern "C" void kernel_launch(void* const* d_in, const int* in_sizes, int n_in,
                              void* d_out, int out_size, void* d_ws, size_t ws_size,
                              hipStream_t stream) {
    (void)in_sizes; (void)n_in; (void)out_size; (void)ws_size;
    const float* output   = (const float*)d_in[0];
    const float* context  = (const float*)d_in[1];
    const float* coverage = (const float*)d_in[2];
    const float* W_w      = (const float*)d_in[3];
    const float* W_b      = (const float*)d_in[4];
    const float* cov_w    = (const float*)d_in[5];
    const float* cov_b    = (const float*)d_in[6];
    const float* out_w    = (const float*)d_in[7];
    const float* out_b    = (const float*)d_in[8];
    float* out = (float*)d_out;
    float* ws  = (float*)d_ws;

    k_scalars<<<B + 1, 256, 0, stream>>>(output, W_w, W_b, cov_w, cov_b, ws);
    k_flash  <<<B * S, 256, FLASH_SMEM_BYTES, stream>>>(context, coverage, W_w, ws, out);
    k_combine<<<B,     256, 0, stream>>>(output, coverage, ws, out);
    k_gemm   <<<dim3(8, 2), 256, 0, stream>>>(out_w, out_b, ws, out);
}